// PrevEmbedding_10866267259469
// MI455X (gfx1250) — compile-verified
//
#include <hip/hip_runtime.h>
#include <stdint.h>

#ifndef __has_builtin
#define __has_builtin(x) 0
#endif

#if defined(__HIP_DEVICE_COMPILE__) && __has_builtin(__builtin_amdgcn_global_load_async_to_lds_b128)
#define USE_ASYNC_LDS 1
#else
#define USE_ASYNC_LDS 0
#endif

namespace {
constexpr int   kH     = 768;        // hidden size (fixed by reference)
constexpr int   kHV    = kH / 4;     // 192 float4 per row
constexpr int   kCPL   = kHV / 32;   // 6 float4 per lane (wave32)
constexpr int   kWaves = 8;          // rows per block (one wave32 per row)
constexpr int   kBlock = 32 * kWaves;
constexpr float kEps   = 1e-5f;
}

typedef float v4f __attribute__((ext_vector_type(4)));   // native 128-bit vector
#if USE_ASYNC_LDS
typedef int v4i __attribute__((ext_vector_type(4)));
typedef __attribute__((address_space(1))) v4i gv4i;      // global int4
typedef __attribute__((address_space(3))) v4i lv4i;      // LDS int4
#endif

// Wave32 butterfly reduction of two running sums (sum, sum-of-squares).
__device__ __forceinline__ void wave_reduce2(float& a, float& b) {
#pragma unroll
  for (int off = 16; off >= 1; off >>= 1) {
    a += __shfl_xor(a, off, 32);
    b += __shfl_xor(b, off, 32);
  }
}

// Grid: (ceil(T/kWaves), B). Each wave32 owns one output row (b, t).
__global__ __launch_bounds__(kBlock) void prev_embed_fused(
    const float* __restrict__ cv,    // [V,H]
    const float* __restrict__ ocr,   // [B,N,H]
    const int*   __restrict__ ids,   // [B,T]
    const float* __restrict__ cvg, const float* __restrict__ cvb,
    const float* __restrict__ og,  const float* __restrict__ ob,
    const float* __restrict__ pos,   // [T,H]
    const float* __restrict__ typ,   // [2,H]
    const float* __restrict__ eg,  const float* __restrict__ eb,
    float* __restrict__ out,         // [B,T,H]
    int T, int V, int N)
{
#if USE_ASYNC_LDS
  __shared__ float stage[kWaves * kH];   // 24 KB of the 320 KB WGP LDS
#endif
  const int lane = threadIdx.x & 31;
  const int wave = threadIdx.x >> 5;
  const int b    = blockIdx.y;
  const int t    = blockIdx.x * kWaves + wave;
  if (t >= T) return;                  // wave-uniform exit
  const int row  = b * T + t;

  // ---- select source row (wave-uniform branch) ----
  const int  id     = ids[row];
  const bool is_ocr = id >= V;
  const int  cvi    = min(max(id, 0), V - 1);
  const int  oci    = min(max(id - V, 0), N - 1);
  const float* src  = is_ocr ? (ocr + ((size_t)b * (size_t)N + (size_t)oci) * kH)
                             : (cv  + (size_t)cvi * kH);
  const float* gam  = is_ocr ? og : cvg;
  const float* bet  = is_ocr ? ob : cvb;

  // gfx1250 prefetch of the pos row we'll need shortly (global_prefetch_b8)
  __builtin_prefetch(pos + (size_t)t * kH + lane * (kH / 32), 0, 3);

  // ---- load the gathered row: 6 x float4 per lane, coalesced ----
  v4f x[kCPL];
#if USE_ASYNC_LDS
  {
    // gfx1250 async global->LDS copy (tracked by ASYNCcnt)
    gv4i* gp = (gv4i*)(uintptr_t)(src + (size_t)lane * 4);                 // lane*16 B
    lv4i* lp = (lv4i*)(unsigned)(uintptr_t)(&stage[wave * kH + lane * 4]); // lane*16 B
#pragma unroll
    for (int j = 0; j < kCPL; ++j)
      __builtin_amdgcn_global_load_async_to_lds_b128(gp + 32 * j, lp + 32 * j, 0, 0);
#if __has_builtin(__builtin_amdgcn_s_wait_asynccnt)
    __builtin_amdgcn_s_wait_asynccnt(0);
#else
    asm volatile("s_wait_asynccnt 0" ::: "memory");
#endif
    const v4f* s4 = reinterpret_cast<const v4f*>(&stage[(size_t)wave * kH]);
#pragma unroll
    for (int j = 0; j < kCPL; ++j) x[j] = s4[lane + 32 * j];
  }
#else
  {
    const v4f* s4 = reinterpret_cast<const v4f*>(src);
#pragma unroll
    for (int j = 0; j < kCPL; ++j) x[j] = s4[lane + 32 * j];
  }
#endif

  // ---- LN stats of the source row ----
  float s = 0.f, sq = 0.f;
#pragma unroll
  for (int j = 0; j < kCPL; ++j) {
    const v4f v = x[j];
    s  += v.x + v.y + v.z + v.w;
    sq += v.x * v.x + v.y * v.y + v.z * v.z + v.w * v.w;
  }
  wave_reduce2(s, sq);
  const float mu   = s * (1.f / kH);
  const float rstd = rsqrtf(sq * (1.f / kH) - mu * mu + kEps);

  // ---- pos + type row, and its LN stats ----
  const int type_id = (t >= V) ? 1 : 0;   // faithful to reference (always 0 here)
  const v4f* pos4 = reinterpret_cast<const v4f*>(pos + (size_t)t * kH);
  const v4f* typ4 = reinterpret_cast<const v4f*>(typ + (size_t)type_id * kH);
  v4f p[kCPL];
  float s2 = 0.f, sq2 = 0.f;
#pragma unroll
  for (int j = 0; j < kCPL; ++j) {
    v4f a = pos4[lane + 32 * j] + typ4[lane + 32 * j];
    p[j] = a;
    s2  += a.x + a.y + a.z + a.w;
    sq2 += a.x * a.x + a.y * a.y + a.z * a.z + a.w * a.w;
  }
  wave_reduce2(s2, sq2);
  const float mu2   = s2 * (1.f / kH);
  const float rstd2 = rsqrtf(sq2 * (1.f / kH) - mu2 * mu2 + kEps);

  // ---- normalize, affine, add, store (non-temporal b128 stores) ----
  const v4f* g4  = reinterpret_cast<const v4f*>(gam);
  const v4f* b4  = reinterpret_cast<const v4f*>(bet);
  const v4f* eg4 = reinterpret_cast<const v4f*>(eg);
  const v4f* eb4 = reinterpret_cast<const v4f*>(eb);
  v4f* out4 = reinterpret_cast<v4f*>(out + (size_t)row * kH);
#pragma unroll
  for (int j = 0; j < kCPL; ++j) {
    const int c = lane + 32 * j;
    // vector math on native ext_vector float4
    const v4f o = (x[j] - mu) * rstd * g4[c] + b4[c]
                + (p[j] - mu2) * rstd2 * eg4[c] + eb4[c];
    __builtin_nontemporal_store(o, out4 + c);   // output is write-once: NT store
  }
}

extern "C" void kernel_launch(void* const* d_in, const int* in_sizes, int n_in,
                              void* d_out, int out_size, void* d_ws, size_t ws_size,
                              hipStream_t stream) {
  const float* cv  = (const float*)d_in[0];   // common_voc_embed [V,H]
  const float* ocr = (const float*)d_in[1];   // ocr_embed        [B,N,H]
  const int*   ids = (const int*)  d_in[2];   // prev_ids         [B,T]
  const float* cvg = (const float*)d_in[3];
  const float* cvb = (const float*)d_in[4];
  const float* og  = (const float*)d_in[5];
  const float* ob  = (const float*)d_in[6];
  const float* pos = (const float*)d_in[7];   // pos_emb [T,H]
  const float* typ = (const float*)d_in[8];   // type_emb [2,H]
  const float* eg  = (const float*)d_in[9];
  const float* eb  = (const float*)d_in[10];
  float* out = (float*)d_out;

  const int H  = in_sizes[3];        // 768
  const int V  = in_sizes[0] / H;    // 32000
  const int T  = in_sizes[7] / H;    // 128
  const int BT = in_sizes[2];        // B*T = 2048
  const int B  = BT / T;             // 16
  const int N  = in_sizes[1] / (B * H);  // 50
  (void)n_in; (void)out_size; (void)d_ws; (void)ws_size; (void)H;

  dim3 grid((T + kWaves - 1) / kWaves, B);
  prev_embed_fused<<<grid, kBlock, 0, stream>>>(cv, ocr, ids, cvg, cvb, og, ob,
                                                pos, typ, eg, eb, out,
                                                T, V, N);
}